// Correlation_10007273800295
// MI455X (gfx1250) — compile-verified
//
#include <hip/hip_runtime.h>

// Problem constants (match reference)
#define Bc 4
#define Cc 64
#define Hc 128
#define Wc 256
#define NDISP 9
#define NBAND 80     // 5 tiles x 16 columns per 16-pixel band
#define PSTR 272     // LDS row stride in float2 units: 2*272 % 64 == 32 -> disjoint bank halves
#define CVSTR 80

typedef __attribute__((ext_vector_type(2))) float v2f;
typedef __attribute__((ext_vector_type(8))) float v8f;

__launch_bounds__(512, 1)
__global__ void corr_wmma_kernel(const float* __restrict__ left,
                                 const float* __restrict__ right,
                                 const float* __restrict__ disp,
                                 const float* __restrict__ occ,
                                 const float* __restrict__ tro,
                                 float* __restrict__ corr)
{
    // Channel-pair interleaved: element [p][x] = { feat[2p][x], feat[2p+1][x] }
    __shared__ float2 sLeft2[(Cc / 2) * PSTR];
    __shared__ float2 sFeat2[(Cc / 2) * PSTR];
    __shared__ float  sCV[16 * 16 * CVSTR];      // per-wave 16x80 correlation bands
    __shared__ float  sTb[Wc];                   // dot(left, trade_off) per pixel

    const int tid = threadIdx.x;
    const int b = blockIdx.x / Hc;
    const int h = blockIdx.x % Hc;

    const size_t plane = (size_t)Hc * Wc;
    const float* lbase = left + (size_t)b * Cc * plane + (size_t)h * Wc;

    // vertical bilinear resample coefficients for this output row
    float iy = fminf(fmaxf((float)h * ((float)Hc / (float)(Hc - 1)) - 0.5f, 0.0f),
                     (float)(Hc - 1));
    float y0f = floorf(iy);
    float wy  = iy - y0f;
    int   y0  = (int)y0f;
    int   y1  = min(y0 + 1, Hc - 1);
    const float* r0 = right + (size_t)b * Cc * plane + (size_t)y0 * Wc;
    const float* r1 = right + (size_t)b * Cc * plane + (size_t)y1 * Wc;

    // ---- Phase 1: stage left row + vertically blended right row into LDS ----
    // One iteration handles a channel pair (2p, 2p+1) x 4 pixels; writes two b128s.
    for (int i = tid; i < (Cc / 2) * (Wc / 4); i += 512) {
        int p  = i >> 6;          // channel pair
        int x4 = (i & 63) << 2;   // pixel in steps of 4
        const size_t o0 = (size_t)(2 * p) * plane + x4;
        const size_t o1 = (size_t)(2 * p + 1) * plane + x4;

        float4 l0 = *(const float4*)(lbase + o0);
        float4 l1 = *(const float4*)(lbase + o1);
        float4 w0 = make_float4(l0.x, l1.x, l0.y, l1.y);
        float4 w1 = make_float4(l0.z, l1.z, l0.w, l1.w);
        *(float4*)&sLeft2[p * PSTR + x4]     = w0;
        *(float4*)&sLeft2[p * PSTR + x4 + 2] = w1;

        float4 a0 = *(const float4*)(r0 + o0);
        float4 b0 = *(const float4*)(r1 + o0);
        float4 a1 = *(const float4*)(r0 + o1);
        float4 b1 = *(const float4*)(r1 + o1);
        float4 f0, f1;
        f0.x = (1.0f - wy) * a0.x + wy * b0.x;
        f0.y = (1.0f - wy) * a0.y + wy * b0.y;
        f0.z = (1.0f - wy) * a0.z + wy * b0.z;
        f0.w = (1.0f - wy) * a0.w + wy * b0.w;
        f1.x = (1.0f - wy) * a1.x + wy * b1.x;
        f1.y = (1.0f - wy) * a1.y + wy * b1.y;
        f1.z = (1.0f - wy) * a1.z + wy * b1.z;
        f1.w = (1.0f - wy) * a1.w + wy * b1.w;
        float4 g0 = make_float4(f0.x, f1.x, f0.y, f1.y);
        float4 g1 = make_float4(f0.z, f1.z, f0.w, f1.w);
        *(float4*)&sFeat2[p * PSTR + x4]     = g0;
        *(float4*)&sFeat2[p * PSTR + x4 + 2] = g1;
    }
    __syncthreads();

    // ---- trade-off dot product: tb[x] = sum_c left[c,x] * trade_off[c,x] ----
    if (tid < Wc) {
        const float* tp = tro + (size_t)b * Cc * plane + (size_t)h * Wc + tid;
        float acc = 0.0f;
        #pragma unroll 8
        for (int p = 0; p < Cc / 2; ++p) {
            float2 v = sLeft2[p * PSTR + tid];
            acc += v.x * tp[(size_t)(2 * p) * plane];
            acc += v.y * tp[(size_t)(2 * p + 1) * plane];
        }
        sTb[tid] = acc;
    }

    // ---- Phase 2: banded row GEMM  CV[x, x'] = sum_c left[c,x] * feat_y[c,x'] ----
    // One wave per 16-pixel M-tile; 5 N-tiles cover x' in [xb-48, xb+32) (clamped).
    const int wave = tid >> 5;
    const int lane = tid & 31;
    const int m    = lane & 15;
    const int half = lane >> 4;    // lanes 0-15 -> K pair p ; lanes 16-31 -> p+1
    const int xb   = wave << 4;
    const int nb0  = min(max(xb - 48, 0), Wc - NBAND);

    v8f accv[5] = {};
    #pragma unroll 4
    for (int k4 = 0; k4 < Cc / 4; ++k4) {
        const int p = (k4 << 1) + half;
        v2f a = *(const v2f*)&sLeft2[p * PSTR + xb + m];     // single ds_load_b64
        #pragma unroll
        for (int t = 0; t < 5; ++t) {
            v2f bv = *(const v2f*)&sFeat2[p * PSTR + nb0 + (t << 4) + m];
            // D = A(16x4 f32) * B(4x16 f32) + C, full fp32 accumulate
            accv[t] = __builtin_amdgcn_wmma_f32_16x16x4_f32(
                false, a, false, bv, (short)0, accv[t], false, false);
        }
    }

    // spill band tiles to LDS for the gather-based epilogue
    {
        float* cvw = &sCV[wave * (16 * CVSTR)];
        const int mrow = half << 3;   // lanes 0-15: M=r ; lanes 16-31: M=r+8
        #pragma unroll
        for (int t = 0; t < 5; ++t) {
            #pragma unroll
            for (int r = 0; r < 8; ++r)
                cvw[(mrow + r) * CVSTR + (t << 4) + m] = accv[t][r];
        }
    }
    __syncthreads();

    // ---- Epilogue: per (pixel, disparity) bilinear gather from CV band ----
    const float* dRow = disp + ((size_t)b * Hc + h) * Wc;
    const float* oRow = occ  + ((size_t)b * Hc + h) * Wc;
    for (int i = tid; i < NDISP * Wc; i += 512) {
        int x = i & (Wc - 1);
        int d = i >> 8;                               // Wc == 256
        float src = (float)x - dRow[x] - (float)(d - (NDISP / 2));
        float ix = fminf(fmaxf(src * ((float)Wc / (float)(Wc - 1)) - 0.5f, 0.0f),
                         (float)(Wc - 1));
        float x0f = floorf(ix);
        float wx  = ix - x0f;
        int x0 = (int)x0f;
        int x1 = min(x0 + 1, Wc - 1);
        int wv = x >> 4;
        int base = min(max((wv << 4) - 48, 0), Wc - NBAND);
        int j0 = min(max(x0 - base, 0), NBAND - 1);
        int j1 = min(max(x1 - base, 0), NBAND - 1);
        const float* cvw = &sCV[wv * (16 * CVSTR) + (x & 15) * CVSTR];
        float v = oRow[x] * ((1.0f - wx) * cvw[j0] + wx * cvw[j1]) + sTb[x];
        corr[(((size_t)b * NDISP + d) * Hc + h) * Wc + x] = v;
    }
}

// 3x3 zero-padded box sum over (B, 9, H, W)
__global__ void box3x3_kernel(const float* __restrict__ src, float* __restrict__ dst)
{
    int idx = blockIdx.x * blockDim.x + threadIdx.x;
    const int total = Bc * NDISP * Hc * Wc;
    if (idx >= total) return;
    int x = idx & (Wc - 1);
    int h = (idx >> 8) & (Hc - 1);
    const float* p = src + (size_t)(idx >> 15) * Hc * Wc;   // (b*9+d) plane
    float acc = 0.0f;
    #pragma unroll
    for (int dy = -1; dy <= 1; ++dy) {
        int hh = h + dy;
        if ((unsigned)hh >= (unsigned)Hc) continue;
        const float* row = p + hh * Wc;
        #pragma unroll
        for (int dx = -1; dx <= 1; ++dx) {
            int xx = x + dx;
            if ((unsigned)xx < (unsigned)Wc) acc += row[xx];
        }
    }
    dst[idx] = acc;
}

extern "C" void kernel_launch(void* const* d_in, const int* in_sizes, int n_in,
                              void* d_out, int out_size, void* d_ws, size_t ws_size,
                              hipStream_t stream)
{
    const float* left  = (const float*)d_in[0];
    const float* right = (const float*)d_in[1];
    const float* disp  = (const float*)d_in[2];
    const float* occ   = (const float*)d_in[3];
    const float* tro   = (const float*)d_in[4];
    float* ws  = (float*)d_ws;     // (B, 9, H, W) pre-boxfilter correlation
    float* out = (float*)d_out;

    corr_wmma_kernel<<<Bc * Hc, 512, 0, stream>>>(left, right, disp, occ, tro, ws);

    const int total = Bc * NDISP * Hc * Wc;
    box3x3_kernel<<<(total + 255) / 256, 256, 0, stream>>>(ws, out);
}